// TorchGemma4VisionPatchEmbedder_49331994362286
// MI455X (gfx1250) — compile-verified
//
#include <hip/hip_runtime.h>
#include <hip/hip_bf16.h>
#include <stdint.h>

typedef __attribute__((ext_vector_type(2))) float v2f;
typedef __attribute__((ext_vector_type(8))) float v8f;
typedef __attribute__((ext_vector_type(4))) int   v4i;

typedef __attribute__((address_space(3))) char       as3_char;
typedef __attribute__((address_space(1))) const char as1_cchar;
typedef __attribute__((address_space(1))) v4i        as1_v4i;   // global int4
typedef __attribute__((address_space(3))) v4i        as3_v4i;   // LDS int4

// Problem constants (from reference setup_inputs)
#define BATCH   16
#define SEQ     4096
#define DIN     588                   // K
#define HID     1152                  // N
#define PTAB    4096                  // P
#define MROWS   (BATCH * SEQ)         // 65536 GEMM rows
#define ROWS_PER_BLOCK 32             // two 16-row M-tiles per block
#define MBLOCKS (MROWS / ROWS_PER_BLOCK)     // 2048
#define WAVES_PER_BLOCK 8             // 8 wave32 = 256 threads
#define NTILES  (HID / 16)            // 72
#define NGROUPS (NTILES / WAVES_PER_BLOCK)   // 9

#if defined(__gfx1250__) && __has_builtin(__builtin_amdgcn_global_load_async_to_lds_b128)
#define HAVE_ASYNC_LDS 1
#else
#define HAVE_ASYNC_LDS 0
#endif

// out[g,c] = 2*(x@W)[g,c] - colsum_W[c] + (pad[g] ? 0 : ptab0[pos0[g],c] + ptab1[pos1[g],c])
__global__ __launch_bounds__(256) void
patch_embed_wmma_f32(const float* __restrict__ x,
                     const int* __restrict__ pos_ids,       // [M,2] int32
                     const unsigned char* __restrict__ pad, // [M] bool
                     const float* __restrict__ W,           // [DIN,HID]
                     const float* __restrict__ ptab,        // [2,PTAB,HID]
                     float* __restrict__ out) {             // [M,HID]
    __shared__ float s_A[ROWS_PER_BLOCK * DIN]; // raw x tile, row stride 588 (conflict-free)
    __shared__ int   s_pos[ROWS_PER_BLOCK * 2]; // clamped position ids
    __shared__ float s_padf[ROWS_PER_BLOCK];    // 0.0 if padded else 1.0

    const int tid      = threadIdx.x;
    const int row_base = blockIdx.x * ROWS_PER_BLOCK;

    // ---- per-row metadata ----
    if (tid < ROWS_PER_BLOCK) {
        const int g  = row_base + tid;
        int p0 = pos_ids[2 * g + 0];
        int p1 = pos_ids[2 * g + 1];
        s_pos[2 * tid + 0] = p0 < 0 ? 0 : p0;
        s_pos[2 * tid + 1] = p1 < 0 ? 0 : p1;
        s_padf[tid] = pad[g] ? 0.0f : 1.0f;
    }

    // ---- stage raw A tile (32 x 588 fp32 = 75264 B) ----
#if HAVE_ASYNC_LDS
    {
        // Async DMA: global -> LDS, 16B per lane per issue, tracked by ASYNCcnt.
        as1_cchar* g1 = (as1_cchar*)(uintptr_t)(const void*)(x + (size_t)row_base * DIN);
        as3_char*  l3 = (as3_char*)(uint32_t)(uintptr_t)(void*)s_A;
        #pragma unroll 2
        for (int i = tid; i < (ROWS_PER_BLOCK * DIN * 4) / 16; i += 256) {
            __builtin_amdgcn_global_load_async_to_lds_b128(
                (as1_v4i*)(g1 + (size_t)i * 16),
                (as3_v4i*)(l3 + i * 16),
                /*offset=*/0, /*cpol=*/0);
        }
#if __has_builtin(__builtin_amdgcn_s_wait_asynccnt)
        __builtin_amdgcn_s_wait_asynccnt(0);
#else
        asm volatile("s_wait_asynccnt 0x0" ::: "memory");
#endif
    }
#else
    {
        const float4* __restrict__ src = (const float4*)(x + (size_t)row_base * DIN);
        float4* dst = (float4*)s_A;
        #pragma unroll 2
        for (int i = tid; i < (ROWS_PER_BLOCK * DIN) / 4; i += 256) {
            dst[i] = src[i];
        }
    }
#endif
    __syncthreads();

    // ---- per-wave: two 16x16 output tiles (rows 0-15 and 16-31), one N-tile ----
    const int wave = tid >> 5;
    const int lane = tid & 31;
    const int half = lane >> 4;     // 0: lanes 0-15, 1: lanes 16-31
    const int rm   = lane & 15;     // A row within tile / B,C,D column index
    const int kpar = half * 2;      // this half-wave holds K = kbase+{0,1} or +{2,3}

    const int col_base = (blockIdx.y * WAVES_PER_BLOCK + wave) * 16;

    const float* __restrict__ A0 = s_A + rm * DIN + kpar;          // tile 0 row
    const float* __restrict__ A1 = s_A + (16 + rm) * DIN + kpar;   // tile 1 row
    const float* __restrict__ Bp = W + (size_t)kpar * HID + col_base + rm;

    v8f acc0 = {};
    v8f acc1 = {};
    float csum = 0.0f;   // partial colsum over this half-wave's k residues

    #pragma unroll 7
    for (int k = 0; k < DIN; k += 4) {
        v2f b;
        b.x = Bp[(size_t)k * HID];
        b.y = Bp[(size_t)k * HID + HID];
        csum += b.x + b.y;
        const v2f a0 = *(const v2f*)(A0 + k);
        const v2f a1 = *(const v2f*)(A1 + k);
        // (neg_a, A, neg_b, B, c_mod, C, reuse_a, reuse_b)
        acc0 = __builtin_amdgcn_wmma_f32_16x16x4_f32(
            false, a0, false, b, (short)0, acc0, false, false);
        acc1 = __builtin_amdgcn_wmma_f32_16x16x4_f32(
            false, a1, false, b, (short)0, acc1, false, false);
    }

    // Lanes L and L^16 hold complementary k residues of column c -> full column sum.
    const float colsum = csum + __shfl_xor(csum, 16, 32);

    // ---- epilogue: affine fold + PE gather ----
    const float* __restrict__ ptab1 = ptab + (size_t)PTAB * HID;
    const int c = col_base + rm;

    #pragma unroll
    for (int r = 0; r < 8; ++r) {
        const int m = r + half * 8;           // tile 0 row
        {
            const int g  = row_base + m;
            const int p0 = s_pos[2 * m + 0];
            const int p1 = s_pos[2 * m + 1];
            const float pe = (ptab[(size_t)p0 * HID + c] + ptab1[(size_t)p1 * HID + c]) * s_padf[m];
            out[(size_t)g * HID + c] = 2.0f * acc0[r] - colsum + pe;
        }
        {
            const int m1 = m + 16;            // tile 1 row
            const int g  = row_base + m1;
            const int p0 = s_pos[2 * m1 + 0];
            const int p1 = s_pos[2 * m1 + 1];
            const float pe = (ptab[(size_t)p0 * HID + c] + ptab1[(size_t)p1 * HID + c]) * s_padf[m1];
            out[(size_t)g * HID + c] = 2.0f * acc1[r] - colsum + pe;
        }
    }
}

extern "C" void kernel_launch(void* const* d_in, const int* in_sizes, int n_in,
                              void* d_out, int out_size, void* d_ws, size_t ws_size,
                              hipStream_t stream) {
    const float*         x    = (const float*)d_in[0];         // pixel_values [16,4096,588] f32
    const int*           pid  = (const int*)d_in[1];           // pixel_position_ids [16,4096,2]
    const unsigned char* pad  = (const unsigned char*)d_in[2]; // padding_positions [16,4096] bool
    const float*         W    = (const float*)d_in[3];         // W_proj [588,1152] f32
    const float*         ptab = (const float*)d_in[4];         // pos_table [2,4096,1152] f32
    float*               out  = (float*)d_out;                 // [16,4096,1152] f32

    dim3 grid(MBLOCKS, NGROUPS);   // 2048 x 9
    dim3 block(256);               // 8 wave32
    patch_embed_wmma_f32<<<grid, block, 0, stream>>>(x, pid, pad, W, ptab, out);
}